// spuLayerTransformer_37503654429000
// MI455X (gfx1250) — compile-verified
//
#include <hip/hip_runtime.h>
#include <hip/hip_bf16.h>

// Elementwise SPU bound propagation: 3 fp32 streams in, 3 fp32 streams out.
// Roofline: 24 B/elem * 16.78M = 402.7 MB -> ~17.3 us at 23.3 TB/s HBM.
// Dataset >> 192 MB L2, so all accesses are non-temporal B128.
// No matmul => WMMA structurally inapplicable; gfx1250 wins used here:
// wave32, SADDR+32-bit-voffset global addressing, B128 NT vmem, and native
// v_exp_f32 / v_rcp_f32 transcendentals (1 op each, <=1 ulp) instead of
// libm expf / IEEE-divide sequences.

typedef __attribute__((ext_vector_type(4))) float v4f;

// sigmoid(-t) = 1/(1+e^t): v_mul (log2e) + v_exp_f32 + v_add + v_rcp_f32.
__device__ __forceinline__ float sig_neg(float t) {
    return __builtin_amdgcn_rcpf(1.0f + __expf(t));
}

__device__ __forceinline__ float fast_rcp(float d) {
    return __builtin_amdgcn_rcpf(d);
}

// spu(t) given precomputed sig = sigmoid(-t)
__device__ __forceinline__ float spu_from_sig(float t, float sig) {
    float pos = __builtin_fmaf(t, t, -0.5f);
    float neg = sig - 1.0f;
    return (t >= 0.0f) ? pos : neg;   // select, like jnp.where
}

__device__ __forceinline__ void spu_elem(float x, float l, float u,
                                         float& xo, float& lo, float& uo) {
    // p = clip(x, l, u)
    float p = fminf(fmaxf(x, l), u);

    float spu_l = spu_from_sig(l, sig_neg(l));
    float spu_u = spu_from_sig(u, sig_neg(u));
    xo          = spu_from_sig(x, sig_neg(x));

    // single sigmoid at p feeds both spu(p) and dx_spu(p)
    float sig_p = sig_neg(p);
    float spu_p = spu_from_sig(p, sig_p);

    // chord through (l, spu(l)) and (u, spu(u))
    float chord_slope = (spu_u - spu_l) * fast_rcp(u - l);
    float chord_int   = __builtin_fmaf(-chord_slope, l, spu_l);

    // tangent at p: dx_spu(p) = p>=0 ? 2p : -sig*(1-sig) = sig^2 - sig
    float tan_slope = (p >= 0.0f) ? (2.0f * p)
                                  : __builtin_fmaf(sig_p, sig_p, -sig_p);
    float tan_int   = __builtin_fmaf(-p, tan_slope, spu_p);

    // line through (l, spu(l)) and (0, -0.5)
    float safe_l   = (l != 0.0f) ? l : 1.0f;
    float l0_slope = (-0.5f - spu_l) * fast_rcp(0.0f - safe_l);
    const float l0_int = -0.5f;

    bool u_pos = (u > 0.0f);
    bool p_nn  = (p >= 0.0f);

    float lb_slope = u_pos ? (p_nn ? tan_slope : l0_slope) : chord_slope;
    float lb_int   = u_pos ? (p_nn ? tan_int   : l0_int)   : chord_int;
    float ub_slope = u_pos ? chord_slope : tan_slope;
    float ub_int   = u_pos ? chord_int   : tan_int;

    // faithful reproduction, including l*(-lb_slope) in u_out's else branch
    lo = lb_int + ((lb_slope > 0.0f) ? l * (-lb_slope) : u * lb_slope);
    uo = ub_int + ((ub_slope > 0.0f) ? u * ub_slope    : l * (-lb_slope));
}

__global__ __launch_bounds__(256)
void spuLayerTransformer_37503654429000_kernel(const float* __restrict__ x,
                                               const float* __restrict__ l,
                                               const float* __restrict__ u,
                                               float* __restrict__ out,
                                               unsigned n) {
    // 32-bit indexing: max byte offset 3n*4 ~ 201 MB < 2^31, so the backend
    // can use scalar-base + 32-bit-voffset global addressing (no 64-bit VALU).
    const unsigned i = (blockIdx.x * blockDim.x + threadIdx.x) * 4u;
    float* __restrict__ out_x = out;
    float* __restrict__ out_l = out + n;
    float* __restrict__ out_u = out + 2u * n;

    if (i + 4u <= n) {
        // 128-bit non-temporal streaming loads (global_load_b128, TH=NT)
        v4f xv = __builtin_nontemporal_load((const v4f*)(x + i));
        v4f lv = __builtin_nontemporal_load((const v4f*)(l + i));
        v4f uv = __builtin_nontemporal_load((const v4f*)(u + i));

        v4f xo, lo, uo;
#pragma unroll
        for (int k = 0; k < 4; ++k) {
            float a, b, c;
            spu_elem(xv[k], lv[k], uv[k], a, b, c);
            xo[k] = a; lo[k] = b; uo[k] = c;
        }

        __builtin_nontemporal_store(xo, (v4f*)(out_x + i));
        __builtin_nontemporal_store(lo, (v4f*)(out_l + i));
        __builtin_nontemporal_store(uo, (v4f*)(out_u + i));
    } else {
        for (unsigned k = i; k < n; ++k) {
            float a, b, c;
            spu_elem(x[k], l[k], u[k], a, b, c);
            out_x[k] = a; out_l[k] = b; out_u[k] = c;
        }
    }
}

extern "C" void kernel_launch(void* const* d_in, const int* in_sizes, int n_in,
                              void* d_out, int out_size, void* d_ws, size_t ws_size,
                              hipStream_t stream) {
    const float* x = (const float*)d_in[0];
    const float* l = (const float*)d_in[1];
    const float* u = (const float*)d_in[2];
    float* out = (float*)d_out;
    const unsigned n = (unsigned)in_sizes[0];   // 16,777,216

    const unsigned nvec   = (n + 3u) / 4u;      // 4 elements per thread
    const unsigned block  = 256u;               // 8 wave32s per block
    const unsigned blocks = (nvec + block - 1u) / block;

    spuLayerTransformer_37503654429000_kernel<<<blocks, block, 0, stream>>>(
        x, l, u, out, n);
}